// Transformer_6416681140458
// MI455X (gfx1250) — compile-verified
//
#include <hip/hip_runtime.h>
#include <hip/hip_bf16.h>
#include <math.h>

// ext-vector types for WMMA operands
typedef __attribute__((ext_vector_type(2))) float v2f;
typedef __attribute__((ext_vector_type(8))) float v8f;

#define EX_PER_WAVE 32
#define WAVES_PER_BLOCK 4
#define EX_PER_BLOCK (EX_PER_WAVE * WAVES_PER_BLOCK)

// ---- minimal complex<float> helpers -------------------------------------
struct cf { float re, im; };
__device__ __forceinline__ cf cmul(cf a, cf b) {
    return { a.re * b.re - a.im * b.im, a.re * b.im + a.im * b.re };
}
__device__ __forceinline__ cf cadd(cf a, cf b) { return { a.re + b.re, a.im + b.im }; }
__device__ __forceinline__ cf cexp_(cf z) {
    float m = __expf(z.re);
    float s, c;
    __sincosf(z.im, &s, &c);
    return { m * c, m * s };
}
__device__ __forceinline__ cf cdiv(cf a, cf b) {
    float inv = 1.0f / (b.re * b.re + b.im * b.im);
    return { (a.re * b.re + a.im * b.im) * inv, (a.im * b.re - a.re * b.im) * inv };
}

// x: (B, 8) f32; Qw/Kw/Vw: (8,4) complex128 flat doubles; Wk: (8,8) c128; Wb: (8,) c128
// out: interleaved (re, im) f32 per example
__global__ __launch_bounds__(EX_PER_BLOCK) void txf_kernel(
    const float* __restrict__ x,
    const double* __restrict__ Qw, const double* __restrict__ Kw,
    const double* __restrict__ Vw,
    const double* __restrict__ Wkw, const double* __restrict__ Wbw,
    float* __restrict__ out, int nex)
{
    // per-wave projection results: [wave][Q/K/V][row(example*2+chunk) 0..63][col 0..15]
    __shared__ float sproj[WAVES_PER_BLOCK][3][64][16];
    __shared__ cf sWk[64];
    __shared__ cf sWb[8];

    const int tid  = threadIdx.x;
    const int w    = tid >> 5;
    const int lane = tid & 31;
    const int blockBase = blockIdx.x * EX_PER_BLOCK;
    const int waveBase  = blockBase + w * EX_PER_WAVE;

    // stage Dense weights (complex128 -> cf) into LDS once per block
    if (tid < 64) {
        sWk[tid] = { (float)Wkw[2 * tid], (float)Wkw[2 * tid + 1] };
    } else if (tid < 72) {
        int i = tid - 64;
        sWb[i] = { (float)Wbw[2 * i], (float)Wbw[2 * i + 1] };
    }

    // speculative prefetch of the next block's x rows (global_prefetch_b8)
    __builtin_prefetch(x + (size_t)(blockBase + EX_PER_BLOCK) * 8 + tid * 8, 0, 1);

    // ---- build B operands (4x16 f32) per lane for Q, K, V ----
    // B[k][n] = (n<8) ? Re(W[n][k]) : Im(W[n-8][k]);  weight flat idx = (e*4+k)*2 + (re?0:1)
    const int ncol = lane & 15;
    const int k0   = (lane < 16) ? 0 : 2;
    const int eidx = ncol & 7;
    const int ri   = ncol >> 3; // 0 = real column, 1 = imag column
    v2f bq, bk, bv;
    bq[0] = (float)Qw[(eidx * 4 + k0) * 2 + ri];
    bq[1] = (float)Qw[(eidx * 4 + k0 + 1) * 2 + ri];
    bk[0] = (float)Kw[(eidx * 4 + k0) * 2 + ri];
    bk[1] = (float)Kw[(eidx * 4 + k0 + 1) * 2 + ri];
    bv[0] = (float)Vw[(eidx * 4 + k0) * 2 + ri];
    bv[1] = (float)Vw[(eidx * 4 + k0 + 1) * 2 + ri];

    // ---- QKV projection via V_WMMA_F32_16X16X4_F32 ----
    // wave handles 32 examples -> 64 (example,chunk) rows -> 4 A-tiles of 16 rows
    const v8f czero = {};
#pragma unroll
    for (int t = 0; t < 4; ++t) {
        // A tile element: lane<16 -> (M=lane&15, K=0,1), lane>=16 -> K=2,3
        int rloc = 16 * t + (lane & 15);     // row within wave's 64 rows
        int eloc = rloc >> 1;                // local example 0..31
        int c    = rloc & 1;                 // chunk 0..1
        int eg   = waveBase + eloc;
        if (eg >= nex) eg = nex - 1;         // clamp address, keep EXEC all-ones
        const float2 av = *(const float2*)(x + (size_t)eg * 8 + c * 4 + k0);
        v2f a; a[0] = av.x; a[1] = av.y;

        v8f dq = __builtin_amdgcn_wmma_f32_16x16x4_f32(
            false, a, false, bq, (short)0, czero, false, false);
        v8f dk = __builtin_amdgcn_wmma_f32_16x16x4_f32(
            false, a, false, bk, (short)0, czero, false, false);
        v8f dv = __builtin_amdgcn_wmma_f32_16x16x4_f32(
            false, a, false, bv, (short)0, czero, false, false);

        // D layout: VGPR j -> row j (lanes 0-15) or row j+8 (lanes 16-31), col = lane&15
        int rowOff = 16 * t + ((lane < 16) ? 0 : 8);
        int colD   = lane & 15;
#pragma unroll
        for (int j = 0; j < 8; ++j) {
            sproj[w][0][rowOff + j][colD] = dq[j];
            sproj[w][1][rowOff + j][colD] = dk[j];
            sproj[w][2][rowOff + j][colD] = dv[j];
        }
    }

    __syncthreads();

    // ---- scalar phase: one thread per example ----
    const int el = lane;                    // local example within wave
    cf Qx[2][8], Kx[2][8], Vx[2][8];        // [chunk][e];  e = h*4 + b
#pragma unroll
    for (int c = 0; c < 2; ++c) {
        int row = el * 2 + c;
#pragma unroll
        for (int e = 0; e < 8; ++e) {
            Qx[c][e] = { sproj[w][0][row][e], sproj[w][0][row][8 + e] };
            Kx[c][e] = { sproj[w][1][row][e], sproj[w][1][row][8 + e] };
            Vx[c][e] = { sproj[w][2][row][e], sproj[w][2][row][8 + e] };
        }
    }

    cf vt[2][8]; // [h][i*4+b]
#pragma unroll
    for (int h = 0; h < 2; ++h) {
        cf att[2][2];
#pragma unroll
        for (int i = 0; i < 2; ++i) {
            cf zn[2];
#pragma unroll
            for (int j = 0; j < 2; ++j) {
                cf acc = { 0.f, 0.f };
#pragma unroll
                for (int b = 0; b < 4; ++b)
                    acc = cadd(acc, cmul(Qx[i][4 * h + b], Kx[j][4 * h + b]));
                zn[j] = { -0.5f * acc.re, -0.5f * acc.im };   // -z, z = dot / sqrt(4)
            }
            float m = fmaxf(zn[0].re, zn[1].re);
            cf e0 = cexp_({ zn[0].re - m, zn[0].im });
            cf e1 = cexp_({ zn[1].re - m, zn[1].im });
            cf s  = cadd(e0, e1);
            att[i][0] = cdiv(e0, s);
            att[i][1] = cdiv(e1, s);
        }
#pragma unroll
        for (int i = 0; i < 2; ++i)
#pragma unroll
            for (int b = 0; b < 4; ++b) {
                cf acc = { 0.f, 0.f };
#pragma unroll
                for (int j = 0; j < 2; ++j)
                    acc = cadd(acc, cmul(att[i][j], Vx[j][4 * h + b]));
                vt[h][i * 4 + b] = acc;
            }
    }

    // Dense gating + gated sum:  sum_{h,m} vt[h][m] * (Wb[m] + sum_l vt[h][l]*Wk[l][m])
    cf total = { 0.f, 0.f };
#pragma unroll
    for (int h = 0; h < 2; ++h) {
#pragma unroll
        for (int m = 0; m < 8; ++m) {
            cf acc = sWb[m];
#pragma unroll
            for (int l = 0; l < 8; ++l)
                acc = cadd(acc, cmul(vt[h][l], sWk[l * 8 + m]));
            total = cadd(total, cmul(vt[h][m], acc));
        }
    }

    int n = blockBase + tid;
    if (n < nex) {
        out[2 * (size_t)n]     = total.re;
        out[2 * (size_t)n + 1] = total.im;
    }
}

extern "C" void kernel_launch(void* const* d_in, const int* in_sizes, int n_in,
                              void* d_out, int out_size, void* d_ws, size_t ws_size,
                              hipStream_t stream) {
    const float*  x   = (const float*)d_in[0];
    const double* Qw  = (const double*)d_in[1];
    const double* Kw  = (const double*)d_in[2];
    const double* Vw  = (const double*)d_in[3];
    const double* Wkw = (const double*)d_in[4];
    const double* Wbw = (const double*)d_in[5];
    float* out = (float*)d_out;

    const int nex  = in_sizes[0] / 8;                 // BATCH
    const int grid = (nex + EX_PER_BLOCK - 1) / EX_PER_BLOCK;
    txf_kernel<<<dim3(grid), dim3(EX_PER_BLOCK), 0, stream>>>(
        x, Qw, Kw, Vw, Wkw, Wbw, out, nex);
}